// GradientDiffUnit_68676527063200
// MI455X (gfx1250) — compile-verified
//
#include <hip/hip_runtime.h>
#include <hip/hip_bf16.h>

typedef __attribute__((ext_vector_type(2))) float v2f;
typedef __attribute__((ext_vector_type(8))) float v8f;

#define BATCH 4
#define CDIM  64
#define NPTS  16384
#define ODIM  128
#define KNN   16
#define NEGINF (-3.0e38f)

__device__ __forceinline__ v8f wmma4(v2f a, v2f b, v8f c) {
  // D = A(16x4) * B(4x16) + C, fp32 in/out
  return __builtin_amdgcn_wmma_f32_16x16x4_f32(
      /*neg_a=*/false, a, /*neg_b=*/false, b,
      /*c_mod=*/(short)0, c, /*reuse_a=*/false, /*reuse_b=*/false);
}

// ---------------------------------------------------------------------------
// Kernel 1: per-point squared norms  norms[b][n] = sum_c x[b][c][n]^2
// ---------------------------------------------------------------------------
__global__ __launch_bounds__(256) void norm_kernel(const float* __restrict__ x,
                                                   float* __restrict__ norms) {
  int i = blockIdx.x * blockDim.x + threadIdx.x;       // 0..B*N-1
  int b = i >> 14;
  int n = i & (NPTS - 1);
  const float* xb = x + (size_t)b * CDIM * NPTS + n;
  float s = 0.0f;
#pragma unroll
  for (int c = 0; c < CDIM; ++c) {
    float v = xb[c * NPTS];
    s += v * v;
  }
  norms[i] = s;
}

// ---------------------------------------------------------------------------
// Kernel 2: KNN via WMMA distance GEMM.
// One wave handles 16 queries (WMMA N dim). Candidates are the M dim.
// Ranking key: 2*inner - cn  (same ordering as -qn - cn + 2*inner).
// The -cn term is folded in as a 17th WMMA (A col = cn, B row = -1).
// Lane L holds query qBase+(L&15); its 8 acc VGPRs are candidates
// candBase + v + 8*(L>>4).  Per-lane register top-16, merged across
// lane pairs (L, L+16) with shfl_xor at the end.
// ---------------------------------------------------------------------------
__global__ __launch_bounds__(256) void knn_kernel(const float* __restrict__ x,
                                                  const float* __restrict__ norms,
                                                  int* __restrict__ idx_out) {
  const int gw   = (blockIdx.x * blockDim.x + threadIdx.x) >> 5; // 0..4095
  const int lane = threadIdx.x & 31;
  const int half = lane >> 4;
  const int l16  = lane & 15;
  const int b     = gw >> 10;            // 1024 query tiles per batch
  const int qBase = (gw & 1023) << 4;
  const float* xb = x + (size_t)b * CDIM * NPTS;
  const float* nb = norms + (size_t)b * NPTS;
  const int q = qBase + l16;

  // Preload query tile as B-operands, pre-scaled by 2 (gives 2*inner).
  // B(4x16) layout: lane holds col N=l16; VGPR0={K=0|K=2}, VGPR1={K=1|K=3}.
  v2f qreg[16];
#pragma unroll
  for (int kk = 0; kk < 16; ++kk) {
    int c = 4 * kk + 2 * half;
    qreg[kk].x = 2.0f * xb[(size_t)c * NPTS + qBase + l16];
    qreg[kk].y = 2.0f * xb[(size_t)(c + 1) * NPTS + qBase + l16];
  }
  // Norm-correction B operand: B[0][n] = -1, rest 0.
  v2f nB;
  nB.x = (half == 0) ? -1.0f : 0.0f;
  nB.y = 0.0f;

  float ld[KNN];
  int   li[KNN];
#pragma unroll
  for (int i = 0; i < KNN; ++i) { ld[i] = NEGINF; li[i] = 0; }
  float listmin = NEGINF;

  for (int candBase = 0; candBase < NPTS; candBase += 16) {
    // A(16x4): candidate tile. Lane holds row M=l16; VGPR0={K=0|K=2}, VGPR1={K=1|K=3}.
    v2f areg[16];
#pragma unroll
    for (int kk = 0; kk < 16; ++kk) {
      int c = 4 * kk + 2 * half;
      areg[kk].x = xb[(size_t)c * NPTS + candBase + l16];
      areg[kk].y = xb[(size_t)(c + 1) * NPTS + candBase + l16];
    }
    float cn = nb[candBase + l16];          // uniform-address-safe for all lanes
    v2f nA;
    nA.x = (half == 0) ? cn : 0.0f;         // A[M][0] = cn[M], other K = 0
    nA.y = 0.0f;

    v8f acc = {};
#pragma unroll
    for (int kk = 0; kk < 16; ++kk) acc = wmma4(areg[kk], qreg[kk], acc);
    acc = wmma4(nA, nB, acc);               // acc = 2*inner - cn

    // Top-16 maintenance (guarded: skip tile if nothing can enter the list).
    float tmax = fmaxf(fmaxf(fmaxf(acc[0], acc[1]), fmaxf(acc[2], acc[3])),
                       fmaxf(fmaxf(acc[4], acc[5]), fmaxf(acc[6], acc[7])));
    if (tmax > listmin) {
      const int cbase = candBase + 8 * half;
#pragma unroll
      for (int v = 0; v < 8; ++v) {
        float s   = acc[v];
        int   cnd = cbase + v;
        if (s > listmin && cnd != q) {      // cnd==q is the masked self-distance
          float cd = s; int ci = cnd;
#pragma unroll
          for (int i = 0; i < KNN; ++i) {   // sorted (desc) insertion network
            bool g = cd > ld[i];
            float td = ld[i]; int ti = li[i];
            ld[i] = g ? cd : td;  li[i] = g ? ci : ti;
            cd    = g ? td : cd;  ci    = g ? ti : ci;
          }
          listmin = ld[KNN - 1];
        }
      }
    }
  }

  // Merge lane L (cands mod16 in [0,8)) with lane L+16 (cands mod16 in [8,16)).
  // Both lists sorted desc: top16(A∪B)[i] = max(A[i], B[15-i]) (order-free downstream).
#pragma unroll
  for (int i = 0; i < KNN; ++i) {
    float od = __shfl_xor(ld[KNN - 1 - i], 16, 32);
    int   oi = __shfl_xor(li[KNN - 1 - i], 16, 32);
    int   ri = (ld[i] >= od) ? li[i] : oi;
    if (lane < 16) idx_out[((size_t)b * NPTS + q) * KNN + i] = ri;
  }
}

// ---------------------------------------------------------------------------
// Kernel 3: z1[b][n][o] = sc[o] * sum_c W1[o][c]   * x[b][c][n]
//           z2[b][n][o] = sc[o] * sum_c (W2-W1)[o][c] * x[b][c][n] + bias[o]
// BN scale/bias folded in.  One wave: fixed (b, o-tile), 8 n-tiles reusing W regs.
// ---------------------------------------------------------------------------
__global__ __launch_bounds__(256) void gemm_kernel(const float* __restrict__ x,
                                                   const float* __restrict__ W,
                                                   const float* __restrict__ gamma,
                                                   const float* __restrict__ beta,
                                                   const float* __restrict__ bn_mean,
                                                   const float* __restrict__ bn_var,
                                                   float* __restrict__ z1,
                                                   float* __restrict__ z2) {
  const int gw   = (blockIdx.x * blockDim.x + threadIdx.x) >> 5; // 0..4095
  const int lane = threadIdx.x & 31;
  const int half = lane >> 4;
  const int l16  = lane & 15;
  const int b     = gw >> 10;
  const int r     = gw & 1023;
  const int otile = r >> 7;       // 0..7
  const int ngrp  = r & 127;      // 0..127
  const int o0    = otile << 4;
  const float* xb = x + (size_t)b * CDIM * NPTS;

  const int o = o0 + l16;
  float sc = gamma[o] * rsqrtf(bn_var[o] + 1e-5f);
  float bs = beta[o] - bn_mean[o] * sc;

  // B-operands: W1^T and (W2-W1)^T slices, scaled by sc (per-lane o).
  v2f w1[16], dw[16];
#pragma unroll
  for (int kk = 0; kk < 16; ++kk) {
    int c = 4 * kk + 2 * half;
    float a0 = W[o * 2 * CDIM + c];
    float a1 = W[o * 2 * CDIM + c + 1];
    float b0 = W[o * 2 * CDIM + CDIM + c];
    float b1 = W[o * 2 * CDIM + CDIM + c + 1];
    w1[kk].x = a0 * sc;        w1[kk].y = a1 * sc;
    dw[kk].x = (b0 - a0) * sc; dw[kk].y = (b1 - a1) * sc;
  }

  for (int t = 0; t < 8; ++t) {
    const int n0 = ((ngrp << 3) + t) << 4;
    v2f areg[16];
#pragma unroll
    for (int kk = 0; kk < 16; ++kk) {
      int c = 4 * kk + 2 * half;
      areg[kk].x = xb[(size_t)c * NPTS + n0 + l16];
      areg[kk].y = xb[(size_t)(c + 1) * NPTS + n0 + l16];
    }
    v8f acc1 = {};
    v8f acc2 = {};
#pragma unroll
    for (int kk = 0; kk < 16; ++kk) acc1 = wmma4(areg[kk], w1[kk], acc1);
#pragma unroll
    for (int kk = 0; kk < 16; ++kk) acc2 = wmma4(areg[kk], dw[kk], acc2);
#pragma unroll
    for (int v = 0; v < 8; ++v) {
      int n = n0 + v + 8 * half;            // D layout: M = vgpr + 8*(lane>>4)
      size_t base = ((size_t)b * NPTS + n) * ODIM + o;
      z1[base] = acc1[v];
      z2[base] = acc2[v] + bs;
    }
  }
}

// ---------------------------------------------------------------------------
// Kernel 4: out[b][o][n] = max_j leaky( z1[b][idx[n][j]][o] + z2[b][n][o] )
// Block = (b, 32 n) x 128 o.  Gathers coalesced over o; output transposed
// through padded LDS for coalesced (B,O,N) stores.
// ---------------------------------------------------------------------------
__global__ __launch_bounds__(256) void epi_kernel(const float* __restrict__ z1,
                                                  const float* __restrict__ z2,
                                                  const int* __restrict__ idx,
                                                  float* __restrict__ out) {
  __shared__ int   sidx[32 * KNN];
  __shared__ float stile[32 * 129];
  const int b  = blockIdx.x >> 9;          // 512 blocks per batch
  const int n0 = (blockIdx.x & 511) << 5;  // 32 n per block
  const int tid = threadIdx.x;

  for (int i = tid; i < 32 * KNN; i += 256)
    sidx[i] = idx[((size_t)b * NPTS + n0) * KNN + i];
  __syncthreads();

  const int o  = tid & 127;
  const int hh = tid >> 7;
  const float* z1b = z1 + (size_t)b * NPTS * ODIM;
  for (int t = 0; t < 16; ++t) {
    int nl = hh * 16 + t;
    int n  = n0 + nl;
    float z2v = z2[((size_t)b * NPTS + n) * ODIM + o];
    float m = NEGINF;
#pragma unroll
    for (int j = 0; j < KNN; ++j) {
      int g = sidx[nl * KNN + j];
      float v = z1b[(size_t)g * ODIM + o] + z2v;
      v = v > 0.0f ? v : 0.2f * v;
      m = fmaxf(m, v);
    }
    stile[nl * 129 + o] = m;
  }
  __syncthreads();

  for (int i = tid; i < 32 * ODIM; i += 256) {
    int o2 = i >> 5;
    int nn = i & 31;
    out[(((size_t)b * ODIM + o2) << 14) + n0 + nn] = stile[nn * 129 + o2];
  }
}

// ---------------------------------------------------------------------------
extern "C" void kernel_launch(void* const* d_in, const int* in_sizes, int n_in,
                              void* d_out, int out_size, void* d_ws, size_t ws_size,
                              hipStream_t stream) {
  const float* x       = (const float*)d_in[0];
  const float* W       = (const float*)d_in[1];
  const float* gamma   = (const float*)d_in[2];
  const float* beta    = (const float*)d_in[3];
  const float* bn_mean = (const float*)d_in[4];
  const float* bn_var  = (const float*)d_in[5];
  float* out = (float*)d_out;

  // Workspace layout (4-byte units): norms | idx | z1 | z2
  float* ws    = (float*)d_ws;
  float* norms = ws;                                        // B*N        = 65536
  int*   idx   = (int*)(ws + BATCH * NPTS);                 // B*N*16     = 1048576
  float* z1    = ws + BATCH * NPTS + BATCH * NPTS * KNN;    // B*N*O      = 8388608
  float* z2    = z1 + (size_t)BATCH * NPTS * ODIM;          // B*N*O

  norm_kernel<<<(BATCH * NPTS) / 256, 256, 0, stream>>>(x, norms);
  knn_kernel<<<512, 256, 0, stream>>>(x, norms, idx);       // 4096 waves
  gemm_kernel<<<512, 256, 0, stream>>>(x, W, gamma, beta, bn_mean, bn_var, z1, z2);
  epi_kernel<<<(BATCH * NPTS) / 32, 256, 0, stream>>>(z1, z2, idx, out);
}